// ForwardSumLoss_5162550690103
// MI455X (gfx1250) — compile-verified
//
#include <hip/hip_runtime.h>

typedef __attribute__((ext_vector_type(2))) float v2f;
typedef __attribute__((ext_vector_type(8))) float v8f;

#define BB 64
#define TT 2000
#define KK 256
#define NEGV (-1e30f)

// ---------------------------------------------------------------------------
// Kernel 1: logZ[b,t] = log( exp(-1) + sum_{k=1..klen} exp(logp[b,t,k-1]) )
// One wave handles 16 rows; sum-of-exps accumulated on the matrix pipe via
// v_wmma_f32_16x16x4_f32 with B = ones.  Branchless masking (select the exp
// INPUT to -1e30 so exp is unconditional -> v_cndmask, no EXEC branches, and
// EXEC stays all-ones around every WMMA).  2000 % 16 == 0 so all 16 rows of a
// wave share one batch index (uniform key_len).
// ---------------------------------------------------------------------------
__global__ __launch_bounds__(256) void logz_kernel(const float* __restrict__ logp,
                                                   const int* __restrict__ key_lens,
                                                   float* __restrict__ logZ) {
    const int lane = threadIdx.x & 31;
    const int wave = blockIdx.x * (blockDim.x >> 5) + (threadIdx.x >> 5);
    const int r0   = wave * 16;                 // 16 rows per wave; 128000 total, no tail
    const int b    = r0 / TT;
    const int klen = key_lens[b];
    const int m    = lane & 15;                 // WMMA M row (0..15)
    const int h    = lane >> 4;                 // half-wave selects K pair {0,1} vs {2,3}
    const float* row = logp + (size_t)(r0 + m) * KK;

    v8f c = {};                                  // f32 accumulator (16x16)
    v2f ones = {1.0f, 1.0f};                     // B matrix: all ones (4x16)

    #pragma unroll 8
    for (int cb = 0; cb < 64; ++cb) {
        const int base = cb * 4 + 2 * h;         // memory column (8B aligned)
        const v2f x = *(const v2f*)(row + base); // one global_load_b64
        const int k0 = base + 1;                 // label index = memcol + 1
        v2f a;
        a.x = __expf((k0     <= klen) ? x.x : NEGV);   // exp(-1e30) == 0
        a.y = __expf((k0 + 1 <= klen) ? x.y : NEGV);
        // D = A(16x4) * ones(4x16) + C : every column of D holds the row sums
        c = __builtin_amdgcn_wmma_f32_16x16x4_f32(false, a, false, ones,
                                                  (short)0, c, false, false);
    }

    // Column N=0 lives in lane 0 (M=0..7 in c[0..7]) and lane 16 (M=8..15).
    const float blank = __expf(-1.0f);
    if ((lane & 15) == 0) {
        const int mbase = (lane >> 4) * 8;
        #pragma unroll
        for (int v = 0; v < 8; ++v)
            logZ[r0 + mbase + v] = __logf(c[v] + blank);
    }
}

// ---------------------------------------------------------------------------
// Kernel 2: CTC forward scan.  One wave32 per batch item; lane l owns states
// [17l, 17l+17) in registers.  Per step: two shfl_up for the neighbor halo,
// in-place descending update (deps point only to lower state indices), no
// block barriers.  Each timestep's 1KB emit row is double-buffered in LDS via
// gfx1250 global_load_async_to_lds_b128 issued one step ahead (ASYNCcnt),
// overlapping the fetch with the 17 logsumexp updates per lane.
// ---------------------------------------------------------------------------
__global__ __launch_bounds__(32) void ctc_scan_kernel(const float* __restrict__ logp,
                                                      const float* __restrict__ logZ,
                                                      const int* __restrict__ key_lens,
                                                      const int* __restrict__ query_lens,
                                                      float* __restrict__ loss_out) {
    __shared__ float rowbuf[2][KK];
    __shared__ float sa[544];

    const int b    = blockIdx.x;
    const int lane = threadIdx.x;               // 0..31, full wave
    const int klen = key_lens[b];
    const int qlen = query_lens[b];
    const float* lp = logp + (size_t)b * TT * KK;
    const float* lz = logZ + (size_t)b * TT;

    // Per-lane addresses for the async row copy: 32 lanes x 16B = 512B per
    // instruction; offset:512 applies to BOTH the LDS and global addresses.
    const unsigned lds_base0 = (unsigned)(size_t)&rowbuf[0][lane * 4];
    const unsigned lds_base1 = (unsigned)(size_t)&rowbuf[1][lane * 4];

    #define ISSUE_ROW_COPY(tcopy, slot)                                          \
        do {                                                                     \
            const float* _g = lp + (size_t)(tcopy) * KK + lane * 4;              \
            unsigned _l = (slot) ? lds_base1 : lds_base0;                        \
            asm volatile(                                                        \
                "global_load_async_to_lds_b128 %0, %1, off\n\t"                  \
                "global_load_async_to_lds_b128 %0, %1, off offset:512"           \
                :: "v"(_l), "v"(_g) : "memory");                                 \
        } while (0)

    float al[17];

    // t = 0: alpha = NEG except states 0 (blank) and 1 (label 1).
    {
        const float lz0 = lz[0];
        const float e1  = lp[0];                // label 1 always <= klen (klen >= 128)
        #pragma unroll
        for (int j = 0; j < 17; ++j) {
            const int s = lane * 17 + j;
            float v = NEGV;
            if (s == 0) v = -1.0f - lz0;
            else if (s == 1) v = e1 - lz0;
            al[j] = v;
        }
    }

    if (qlen > 1) ISSUE_ROW_COPY(1, 1);

    for (int t = 1; t < qlen; ++t) {
        const int  cur  = t & 1;
        const bool more = (t + 1) < qlen;

        // WAR guard: previous step's ds_loads of the other buffer must retire
        // before the async engine overwrites it (async vs DS are unordered).
        asm volatile("s_wait_dscnt 0x0" ::: "memory");
        if (more) {
            ISSUE_ROW_COPY(t + 1, cur ^ 1);
            asm volatile("s_wait_asynccnt 0x2" ::: "memory");  // row t ready
        } else {
            asm volatile("s_wait_asynccnt 0x0" ::: "memory");
        }

        const float lzt   = lz[t];
        const float blank = -1.0f - lzt;

        float up1 = __shfl_up(al[16], 1);        // alpha[17l - 1]
        float up2 = __shfl_up(al[15], 1);        // alpha[17l - 2]
        up1 = lane ? up1 : NEGV;                 // branchless lane-0 fixup
        up2 = lane ? up2 : NEGV;

        #pragma unroll
        for (int j = 16; j >= 0; --j) {
            const int s  = lane * 17 + j;
            const float a0 = al[j];
            const float a1 = (j == 0) ? up1 : al[j - 1];
            float       a2 = (j == 0) ? up2 : ((j == 1) ? up1 : al[j - 2]);
            const bool odd   = (s & 1) != 0;
            const int  labm1 = (s - 1) >> 1;     // label-1 (valid when odd)
            const int  col   = min(max(s - 1, 0) >> 1, KK - 1);
            if (!(odd && labm1 >= 1)) a2 = NEGV; // skip only for odd s >= 3
            const float raw  = rowbuf[cur][col]; // unconditional LDS read
            const float oemt = (labm1 < klen) ? (raw - lzt) : NEGV;
            const float emit = odd ? oemt : blank;
            const float mx = fmaxf(a0, fmaxf(a1, a2));
            const float sm = __expf(a0 - mx) + __expf(a1 - mx) + __expf(a2 - mx);
            al[j] = mx + __logf(sm) + emit;
        }
    }

    // Gather final two states via LDS (single wave).
    #pragma unroll
    for (int j = 0; j < 17; ++j) sa[lane * 17 + j] = al[j];
    __syncthreads();
    if (lane == 0) {
        const int   last = 2 * klen;              // <= 512
        const float a = sa[last], p = sa[last - 1];
        const float mx = fmaxf(a, p);
        const float nll = -(mx + __logf(__expf(a - mx) + __expf(p - mx)));
        const float l   = nll / (float)(klen > 1 ? klen : 1);
        loss_out[b] = (nll > 5e29f) ? 0.0f : l;
    }
    #undef ISSUE_ROW_COPY
}

// ---------------------------------------------------------------------------
// Kernel 3: mean over the 64 per-batch losses -> d_out[0].
// ---------------------------------------------------------------------------
__global__ __launch_bounds__(32) void finalize_kernel(const float* __restrict__ loss,
                                                      float* __restrict__ out) {
    const int lane = threadIdx.x;
    float v = loss[lane] + loss[lane + 32];
    #pragma unroll
    for (int off = 16; off > 0; off >>= 1) v += __shfl_down(v, off);
    if (lane == 0) out[0] = v / (float)BB;
}

extern "C" void kernel_launch(void* const* d_in, const int* in_sizes, int n_in,
                              void* d_out, int out_size, void* d_ws, size_t ws_size,
                              hipStream_t stream) {
    const float* attn       = (const float*)d_in[0];   // (B,1,T,K) f32
    const int*   key_lens   = (const int*)d_in[1];     // (B,)
    const int*   query_lens = (const int*)d_in[2];     // (B,)
    float* logZ = (float*)d_ws;                        // B*T floats
    float* loss = logZ + (size_t)BB * TT;              // B floats
    float* out  = (float*)d_out;

    // 8000 waves * 16 rows = 128000 rows, 8 waves per block -> 1000 blocks.
    logz_kernel<<<1000, 256, 0, stream>>>(attn, key_lens, logZ);
    ctc_scan_kernel<<<BB, 32, 0, stream>>>(attn, logZ, key_lens, query_lens, loss);
    finalize_kernel<<<1, 32, 0, stream>>>(loss, out);
}